// MultiHeadAttention_45294725104087
// MI455X (gfx1250) — compile-verified
//
#include <hip/hip_runtime.h>
#include <math.h>

typedef __attribute__((ext_vector_type(2))) float v2f;
typedef __attribute__((ext_vector_type(8))) float v8f;

#define B_  2
#define S_  2048
#define D_  1024
#define H_  16
#define DK_ 64

__device__ __forceinline__ v2f ld2g(const float* p) { return *(const v2f*)p; }

// ---------------------------------------------------------------------------
// Kernel 1: fused QKV projection.
// Q = (x @ Wq^T + bq) * 1/sqrt(DK), K = x @ Wk^T + bk, V = x @ Wv^T + bv
// Output layout: [b, h, s, dk] (head-major) for attention consumption.
// Each wave computes one 16(M) x 16(N) tile; A-frag of x shared by 3 WMMAs.
// ---------------------------------------------------------------------------
__global__ void qkv_proj_kernel(const float* __restrict__ x,
                                const float* __restrict__ Wq, const float* __restrict__ bq,
                                const float* __restrict__ Wk, const float* __restrict__ bk,
                                const float* __restrict__ Wv, const float* __restrict__ bv,
                                float* __restrict__ Qo, float* __restrict__ Ko,
                                float* __restrict__ Vo)
{
    const int lane = threadIdx.x & 31;
    const int wave = threadIdx.x >> 5;
    const int l16  = lane & 15;
    const int half = lane >> 4;
    const int koff = 2 * half;

    const int mtile = blockIdx.x;              // 0..255 (B*S/16)
    const int ntile = blockIdx.y * 4 + wave;   // 0..63  (D/16)
    const int row0  = mtile * 16;              // flattened (b,s) row
    const int col0  = ntile * 16;              // output feature j

    const float* xrow  = x  + (size_t)(row0 + l16) * D_;   // A: m = l16
    const float* wqrow = Wq + (size_t)(col0 + l16) * D_;   // B: n = l16 (W^T access)
    const float* wkrow = Wk + (size_t)(col0 + l16) * D_;
    const float* wvrow = Wv + (size_t)(col0 + l16) * D_;

    v8f accq = {}; v8f acck = {}; v8f accv = {};
    for (int k0 = 0; k0 < D_; k0 += 4) {
        v2f a  = ld2g(xrow  + k0 + koff);
        v2f b0 = ld2g(wqrow + k0 + koff);
        v2f b1 = ld2g(wkrow + k0 + koff);
        v2f b2 = ld2g(wvrow + k0 + koff);
        accq = __builtin_amdgcn_wmma_f32_16x16x4_f32(false, a, false, b0, (short)0, accq, false, false);
        acck = __builtin_amdgcn_wmma_f32_16x16x4_f32(false, a, false, b1, (short)0, acck, false, false);
        accv = __builtin_amdgcn_wmma_f32_16x16x4_f32(false, a, false, b2, (short)0, accv, false, false);
    }

    const float bqv = bq[col0 + l16];
    const float bkv = bk[col0 + l16];
    const float bvv = bv[col0 + l16];
    const float qscale = 0.125f;               // 1/sqrt(64)

    const int h    = col0 >> 6;                // head index
    const int dk0  = col0 & (DK_ - 1);
    const int b    = row0 / S_;
    const int s0   = row0 % S_;
    const size_t headbase = (size_t)(b * H_ + h) * S_;

#pragma unroll
    for (int r = 0; r < 8; ++r) {
        const int m = r + 8 * half;
        const size_t idx = (headbase + (size_t)(s0 + m)) * DK_ + dk0 + l16;
        Qo[idx] = (accq[r] + bqv) * qscale;
        Ko[idx] = acck[r] + bkv;
        Vo[idx] = accv[r] + bvv;
    }
}

// ---------------------------------------------------------------------------
// Kernel 2: attention for one (b, h, 16-query strip).
// 4 waves / 128 threads per block. LDS holds the full 16 x 2048 score strip
// (128 KB < 320 KB WGP LDS). Phases:
//   (1) scores = Q K^T via WMMA (Q pre-scaled), mask applied, into LDS
//   (2) row softmax in LDS
//   (3) coalesced normalized-weights write to global
//   (4) P @ V via WMMA with A-frag streamed from LDS, ctx written [b,s,d]
// ---------------------------------------------------------------------------
__global__ void attention_kernel(const float* __restrict__ Qm,
                                 const float* __restrict__ Km,
                                 const float* __restrict__ Vm,
                                 const int*   __restrict__ mask,
                                 float* __restrict__ weights,
                                 float* __restrict__ ctx)
{
    extern __shared__ float sc[];              // [16][S_] score strip
    __shared__ float red[16 * 8];
    __shared__ float rowmax[16];
    __shared__ float rowinv[16];

    const int tid  = threadIdx.x;
    const int lane = tid & 31;
    const int wave = tid >> 5;                 // 0..3
    const int l16  = lane & 15;
    const int half = lane >> 4;
    const int koff = 2 * half;

    const int bh = blockIdx.x;                 // 0..31  (b*H + h)
    const int b  = bh / H_;
    const int h  = bh % H_;
    const int q0 = blockIdx.y * 16;            // query strip base

    const size_t headbase = (size_t)bh * S_;

    // Preload Q A-fragments for the strip: qa[t] covers dk = 4t..4t+3.
    v2f qa[16];
    {
        const float* qrow = Qm + (headbase + q0 + l16) * DK_;
#pragma unroll
        for (int t = 0; t < 16; ++t) qa[t] = ld2g(qrow + 4 * t + koff);
    }

    // ---- Phase 1: scores -> LDS --------------------------------------------
    for (int kt = wave; kt < S_ / 16; kt += 4) {
        const int key0 = kt * 16;
        const float* krow = Km + (headbase + key0 + l16) * DK_;  // B: n = l16
        v8f acc = {};
#pragma unroll
        for (int t = 0; t < 16; ++t) {
            v2f bf = ld2g(krow + 4 * t + koff);
            acc = __builtin_amdgcn_wmma_f32_16x16x4_f32(false, qa[t], false, bf, (short)0, acc, false, false);
        }
#pragma unroll
        for (int r = 0; r < 8; ++r) {
            const int m = r + 8 * half;
            const int mk = mask[((size_t)b * S_ + (q0 + m)) * S_ + key0 + l16];
            sc[m * S_ + key0 + l16] = (mk == 0) ? -INFINITY : acc[r];
        }
    }
    __syncthreads();

    // ---- Phase 2: row softmax (8 threads per row) --------------------------
    const int row = tid >> 3;                  // 0..15
    const int sub = tid & 7;                   // 0..7
    float mx = -INFINITY;
    for (int c = sub; c < S_; c += 8) mx = fmaxf(mx, sc[row * S_ + c]);
    red[row * 8 + sub] = mx;
    __syncthreads();
    if (sub == 0) {
        float m2 = red[row * 8];
#pragma unroll
        for (int i = 1; i < 8; ++i) m2 = fmaxf(m2, red[row * 8 + i]);
        rowmax[row] = m2;
    }
    __syncthreads();
    const float rm = rowmax[row];
    float sum = 0.f;
    for (int c = sub; c < S_; c += 8) {
        const float e = __expf(sc[row * S_ + c] - rm);
        sc[row * S_ + c] = e;
        sum += e;
    }
    red[row * 8 + sub] = sum;
    __syncthreads();
    if (sub == 0) {
        float s2 = 0.f;
#pragma unroll
        for (int i = 0; i < 8; ++i) s2 += red[row * 8 + i];
        rowinv[row] = 1.0f / s2;
    }
    __syncthreads();

    // ---- Phase 3: normalize + coalesced weights write ----------------------
    float* wout = weights + (headbase + q0) * S_;
    for (int idx = tid; idx < 16 * S_; idx += 128) {
        const int r2 = idx >> 11;              // idx / S_
        const float w = sc[idx] * rowinv[r2];
        sc[idx]   = w;
        wout[idx] = w;                         // row stride == S_, contiguous
    }
    __syncthreads();

    // ---- Phase 4: ctx tile = P @ V  (each wave: 16 q x 16 dk) --------------
    {
        const int dk0 = wave * 16;
        v8f acc = {};
        for (int kk = 0; kk < S_; kk += 4) {
            const v2f a = *(const v2f*)(&sc[l16 * S_ + kk + koff]);   // A: m = l16
            const float* vp = Vm + (headbase + kk + koff) * DK_ + dk0 + l16;
            v2f bf = { vp[0], vp[DK_] };                               // B[k][n], k=koff,koff+1
            acc = __builtin_amdgcn_wmma_f32_16x16x4_f32(false, a, false, bf, (short)0, acc, false, false);
        }
#pragma unroll
        for (int r = 0; r < 8; ++r) {
            const int m = r + 8 * half;
            ctx[((size_t)(b * S_ + q0 + m)) * D_ + h * DK_ + dk0 + l16] = acc[r];
        }
    }
}

// ---------------------------------------------------------------------------
// Kernel 3: output projection  out = ctx @ Wo^T + bo
// ---------------------------------------------------------------------------
__global__ void out_proj_kernel(const float* __restrict__ ctx,
                                const float* __restrict__ Wo,
                                const float* __restrict__ bo,
                                float* __restrict__ out)
{
    const int lane = threadIdx.x & 31;
    const int wave = threadIdx.x >> 5;
    const int l16  = lane & 15;
    const int half = lane >> 4;
    const int koff = 2 * half;

    const int mtile = blockIdx.x;
    const int ntile = blockIdx.y * 4 + wave;
    const int row0  = mtile * 16;
    const int col0  = ntile * 16;

    const float* arow = ctx + (size_t)(row0 + l16) * D_;
    const float* wrow = Wo  + (size_t)(col0 + l16) * D_;

    v8f acc = {};
    for (int k0 = 0; k0 < D_; k0 += 4) {
        v2f a  = ld2g(arow + k0 + koff);
        v2f bf = ld2g(wrow + k0 + koff);
        acc = __builtin_amdgcn_wmma_f32_16x16x4_f32(false, a, false, bf, (short)0, acc, false, false);
    }

    const float bv = bo[col0 + l16];
#pragma unroll
    for (int r = 0; r < 8; ++r) {
        const int m = r + 8 * half;
        out[(size_t)(row0 + m) * D_ + col0 + l16] = acc[r] + bv;
    }
}

// ---------------------------------------------------------------------------
extern "C" void kernel_launch(void* const* d_in, const int* in_sizes, int n_in,
                              void* d_out, int out_size, void* d_ws, size_t ws_size,
                              hipStream_t stream) {
    const float* x    = (const float*)d_in[0];
    const int*   mask = (const int*)  d_in[1];
    const float* Wq   = (const float*)d_in[2];
    const float* bq   = (const float*)d_in[3];
    const float* Wk   = (const float*)d_in[4];
    const float* bk   = (const float*)d_in[5];
    const float* Wv   = (const float*)d_in[6];
    const float* bv   = (const float*)d_in[7];
    const float* Wo   = (const float*)d_in[8];
    const float* bo   = (const float*)d_in[9];

    float* out     = (float*)d_out;
    float* weights = out + (size_t)B_ * S_ * D_;   // tuple output #2

    const size_t headElems = (size_t)B_ * H_ * S_ * DK_;   // 4,194,304
    float* Qm  = (float*)d_ws;
    float* Km  = Qm + headElems;
    float* Vm  = Km + headElems;
    float* ctx = Vm + headElems;                           // total 64 MB

    // 1) fused QKV projection: 256 M-tiles x 64 N-tiles, 4 waves/block
    qkv_proj_kernel<<<dim3(B_ * S_ / 16, D_ / 16 / 4), 128, 0, stream>>>(
        x, Wq, bq, Wk, bk, Wv, bv, Qm, Km, Vm);

    // 2) attention: one block per (b*h, 16-query strip), 128 KB dynamic LDS
    attention_kernel<<<dim3(B_ * H_, S_ / 16), 128, 16 * S_ * sizeof(float), stream>>>(
        Qm, Km, Vm, mask, weights, ctx);

    // 3) output projection
    out_proj_kernel<<<dim3(B_ * S_ / 16, D_ / 16 / 4), 128, 0, stream>>>(
        ctx, Wo, bo, out);
}